// RoiPooling_65154653880572
// MI455X (gfx1250) — compile-verified
//
#include <hip/hip_runtime.h>

// ROI bilinear pooling (ROI-align style), NHWC image (1, 512, 512, 256) f32.
// out[roi, pr, pc, ch] = bilinear blend of 4 corner pixels, separable weights.
// Memory-bound (~75 MB traffic, ~30 MFLOP) -> optimize data movement:
//  - block per (roi, pr), thread == channel -> fully coalesced 1KB bursts
//  - gfx1250 async-to-LDS staging of the 4 corner rows, double buffered
//    across the 7 pool columns (ASYNCcnt + barrier sync, CDNA5 path).

#define POOL 7
#define NUM_ROIS 300
#define IMG 512
#define NCH 256

#if defined(__gfx1250__) &&                                                    \
    __has_builtin(__builtin_amdgcn_global_load_async_to_lds_b128) &&           \
    __has_builtin(__builtin_amdgcn_s_wait_asynccnt)
#define USE_ASYNC_LDS 1
#else
#define USE_ASYNC_LDS 0
#endif

__global__ __launch_bounds__(256) void roi_bilinear_kernel(
    const float* __restrict__ img,   // (512, 512, 256) flat
    const int* __restrict__ rois,    // (300, 4) flat: x, y, w, h
    float* __restrict__ out)         // (300, 7, 7, 256) flat
{
    const int bid = blockIdx.x;
    const int roi = bid / POOL;
    const int pr  = bid % POOL;
    const int t   = threadIdx.x;     // channel index (and stage chunk id)

    const int x = rois[roi * 4 + 0];
    const int y = rois[roi * 4 + 1];
    const int w = rois[roi * 4 + 2];
    const int h = rois[roi * 4 + 3];

    // Row coordinates for this pool row (matches reference _axis_coords, f32 math).
    const float scale_r = (float)w / 7.0f;
    const float src_r   = (float)pr * scale_r;
    const int   lo_r    = (int)floorf(src_r);
    const int   r0      = x + lo_r;
    const int   r1      = x + min(lo_r + 1, w - 1);
    const float fr      = src_r - (float)lo_r;

    // Column coordinates for all 7 pool columns.
    int   c0[POOL], c1[POOL];
    float fc[POOL];
    const float scale_c = (float)h / 7.0f;
#pragma unroll
    for (int pc = 0; pc < POOL; ++pc) {
        const float src = (float)pc * scale_c;
        const int   lo  = (int)floorf(src);
        c0[pc] = y + lo;
        c1[pc] = y + min(lo + 1, h - 1);
        fc[pc] = src - (float)lo;
    }

    float* outBase = out + (size_t)(roi * POOL + pr) * POOL * NCH;

#if USE_ASYNC_LDS
    // Double-buffered stage of 4 corner pixel vectors (4 x 256 f32 = 4KB/stage).
    __shared__ __attribute__((aligned(16))) float buf[2][4 * NCH];

    typedef int v4i __attribute__((ext_vector_type(4)));
    typedef __attribute__((address_space(1))) v4i g_v4i;   // global source
    typedef __attribute__((address_space(3))) v4i l_v4i;   // LDS destination

    const int p    = t >> 6;                 // corner pixel 0..3
    const int q    = t & 63;                 // 16B chunk within the 1KB pixel
    const int prow = (p < 2) ? r0 : r1;      // corners 0,1 use r0; 2,3 use r1

    auto stage = [&](int pc, int b) {
        const int    pcol = (p & 1) ? c1[pc] : c0[pc];
        const float* g    = img + (size_t)(prow * IMG + pcol) * NCH + q * 4;
        // Each lane copies 16B global->LDS; one instruction per wave per stage.
        __builtin_amdgcn_global_load_async_to_lds_b128(
            (g_v4i*)g, (l_v4i*)&buf[b][p * NCH + q * 4], 0, 0);
    };

    stage(0, 0);
#pragma unroll
    for (int pc = 0; pc < POOL; ++pc) {
        __builtin_amdgcn_s_wait_asynccnt(0);  // this wave's stage(pc) landed in LDS
        __syncthreads();                      // all waves' chunks visible; prior reads done
        if (pc + 1 < POOL) stage(pc + 1, (pc + 1) & 1);  // overlap next gather

        const int   b   = pc & 1;
        const float f   = fc[pc];
        const float v00 = buf[b][0 * NCH + t];
        const float v01 = buf[b][1 * NCH + t];
        const float v10 = buf[b][2 * NCH + t];
        const float v11 = buf[b][3 * NCH + t];
        const float top = v00 * (1.0f - f) + v01 * f;
        const float bot = v10 * (1.0f - f) + v11 * f;
        outBase[pc * NCH + t] = top * (1.0f - fr) + bot * fr;
    }
#else
    // Fallback: direct coalesced loads (thread == channel).
    const float* rowA = img + (size_t)r0 * IMG * NCH;
    const float* rowB = img + (size_t)r1 * IMG * NCH;
#pragma unroll
    for (int pc = 0; pc < POOL; ++pc) {
        const float f   = fc[pc];
        const float v00 = rowA[(size_t)c0[pc] * NCH + t];
        const float v01 = rowA[(size_t)c1[pc] * NCH + t];
        const float v10 = rowB[(size_t)c0[pc] * NCH + t];
        const float v11 = rowB[(size_t)c1[pc] * NCH + t];
        const float top = v00 * (1.0f - f) + v01 * f;
        const float bot = v10 * (1.0f - f) + v11 * f;
        outBase[pc * NCH + t] = top * (1.0f - fr) + bot * fr;
    }
#endif
}

extern "C" void kernel_launch(void* const* d_in, const int* in_sizes, int n_in,
                              void* d_out, int out_size, void* d_ws, size_t ws_size,
                              hipStream_t stream) {
    const float* img  = (const float*)d_in[0];
    const int*   rois = (const int*)d_in[1];
    float*       out  = (float*)d_out;
    (void)in_sizes; (void)n_in; (void)out_size; (void)d_ws; (void)ws_size;

    dim3 grid(NUM_ROIS * POOL);   // 2100 blocks: one per (roi, pool_row)
    dim3 block(256);              // thread == channel; 8 wave32s per block
    roi_bilinear_kernel<<<grid, block, 0, stream>>>(img, rois, out);
}